// CRFLoss_67104569033151
// MI455X (gfx1250) — compile-verified
//
#include <hip/hip_runtime.h>

// ---------------------------------------------------------------------------
// Dense-CRF relaxed Potts loss on MI455X (gfx1250).
// Gram matrix of 5-D bilateral features via V_WMMA_F32_16X16X4_F32 (K padded
// to 8 => two WMMAs per 16x16 tile), epilogue (exp + weighting) in VALU which
// co-executes with the XDL WMMA pipe. All data (~370 KB) is L2-resident.
// ---------------------------------------------------------------------------

typedef float v2f __attribute__((ext_vector_type(2)));
typedef float v8f __attribute__((ext_vector_type(8)));

constexpr int   kH = 96, kW = 96;
constexpr int   kN = kH * kW;                  // 9216 pixels
constexpr float kInvSigmaXY  = 1.0f / 15.0f;
constexpr float kInvSigmaRGB = 8.0f;           // 1 / 0.125
constexpr int   kTiles = kN / 16;              // 576 16-row/col tiles
constexpr int   kWavesPerBlock = 8;            // 256 threads = 8 wave32
constexpr int   kBlocks = kTiles / kWavesPerBlock; // 72
// exp(-0.5*d2) == exp2(d2 * (-0.5*log2(e)))
constexpr float kExpScale = -0.72134752044448170367f;

// ---------------------------------------------------------------------------
// Phase 1: build padded features feat[N][8] = {x/15, y/15, 8r, 8g, 8b, 0,0,0},
// squared norms sq[N], and seg-prob copy sval[N].
// ---------------------------------------------------------------------------
__global__ void crf_prep(const float* __restrict__ s,
                         const float* __restrict__ img,
                         float* __restrict__ feat,
                         float* __restrict__ sq,
                         float* __restrict__ sval) {
  int p = blockIdx.x * blockDim.x + threadIdx.x;
  if (p >= kN) return;
  float x = (float)(p % kW) * kInvSigmaXY;     // pos = (xx, yy)/15, n = y*W + x
  float y = (float)(p / kW) * kInvSigmaXY;
  float r = img[p]          * kInvSigmaRGB;
  float g = img[kN + p]     * kInvSigmaRGB;
  float b = img[2 * kN + p] * kInvSigmaRGB;
  float4* f4 = (float4*)feat;
  f4[p * 2 + 0] = make_float4(x, y, r, g);     // K = 0..3
  f4[p * 2 + 1] = make_float4(b, 0.0f, 0.0f, 0.0f); // K = 4..7 (zero padded)
  sq[p]   = x * x + y * y + r * r + g * g + b * b;
  sval[p] = s[p];
}

// ---------------------------------------------------------------------------
// Phase 2: one wave per 16-row i-tile, sweep all 576 j-tiles.
//   G = A(16x8) * B(8x16) via two v_wmma_f32_16x16x4_f32
//   d2 = sqI + sqJ - 2G ;  w = exp(-0.5*max(d2,0)) ; acc += w*s_i*(1-s_j)
// f32 A fragment (16x4): lane L holds M = L&15, K0 = (L>>4)*2 in 2 VGPRs.
// f32 C/D (16x16):       VGPR v holds M = v + 8*(L>=16), N = L&15.
// ---------------------------------------------------------------------------
__global__ void __launch_bounds__(256) crf_tiles(
    const float* __restrict__ feat, const float* __restrict__ sq,
    const float* __restrict__ sval, float* __restrict__ partials) {
  const int tid  = threadIdx.x;
  const int wave = tid >> 5;
  const int lane = tid & 31;
  const int half = lane >> 4;    // 0: lanes 0-15, 1: lanes 16-31
  const int nm   = lane & 15;    // M (for A) / N (for B,C,D)
  const int i0   = (blockIdx.x * kWavesPerBlock + wave) * 16;
  const int K0   = half * 2;

  // Loop-invariant A fragments + per-M scalars for this i-tile.
  const int arow = i0 + nm;
  v2f a0 = *(const v2f*)(feat + arow * 8 + K0);       // K = K0, K0+1
  v2f a1 = *(const v2f*)(feat + arow * 8 + K0 + 4);   // K = K0+4, K0+5
  float sqI[8], sI[8];
#pragma unroll
  for (int v = 0; v < 8; ++v) {
    int M = v + half * 8;        // matches C/D VGPR layout
    sqI[v] = sq[i0 + M];
    sI[v]  = sval[i0 + M];
  }

  float acc = 0.0f;
  for (int jt = 0; jt < kTiles; ++jt) {
    const int col = jt * 16 + nm;
    v2f b0 = *(const v2f*)(feat + col * 8 + K0);      // B fragment, column = N
    v2f b1 = *(const v2f*)(feat + col * 8 + K0 + 4);
    float sqJ = sq[col];                              // per-lane N = lane&15
    float bJ  = 1.0f - sval[col];
    __builtin_prefetch(feat + (col + 16) * 8, 0, 1);  // next j-tile -> L1/L2

    v8f c = {};
    // D = A*B + C ; (neg_a, A, neg_b, B, c_mod, C, reuse_a, reuse_b)
    c = __builtin_amdgcn_wmma_f32_16x16x4_f32(false, a0, false, b0,
                                              (short)0, c, false, false);
    c = __builtin_amdgcn_wmma_f32_16x16x4_f32(false, a1, false, b1,
                                              (short)0, c, false, false);

    float rowsum = 0.0f;
#pragma unroll
    for (int v = 0; v < 8; ++v) {
      float d2 = fmaxf(sqI[v] + sqJ - 2.0f * c[v], 0.0f);
      float w  = __builtin_amdgcn_exp2f(kExpScale * d2); // v_exp_f32
      rowsum   = fmaf(w, sI[v], rowsum);                 // * s_i
    }
    acc = fmaf(rowsum, bJ, acc);                         // * (1 - s_j)
  }

  // Deterministic reduction: wave32 butterfly -> LDS -> per-block partial.
#pragma unroll
  for (int off = 16; off >= 1; off >>= 1)
    acc += __shfl_xor(acc, off, 32);

  __shared__ float wsum[kWavesPerBlock];
  if (lane == 0) wsum[wave] = acc;
  __syncthreads();
  if (tid == 0) {
    float t = 0.0f;
#pragma unroll
    for (int w = 0; w < kWavesPerBlock; ++w) t += wsum[w];
    partials[blockIdx.x] = t;
  }
}

// ---------------------------------------------------------------------------
// Phase 3: fixed-order final sum (bit-deterministic), normalize by N.
// ---------------------------------------------------------------------------
__global__ void crf_finish(const float* __restrict__ partials,
                           float* __restrict__ out) {
  if (threadIdx.x == 0 && blockIdx.x == 0) {
    float t = 0.0f;
    for (int i = 0; i < kBlocks; ++i) t += partials[i];
    out[0] = t / (float)kN;
  }
}

extern "C" void kernel_launch(void* const* d_in, const int* in_sizes, int n_in,
                              void* d_out, int out_size, void* d_ws,
                              size_t ws_size, hipStream_t stream) {
  (void)in_sizes; (void)n_in; (void)out_size; (void)ws_size;
  const float* s   = (const float*)d_in[0];   // [1,1,96,96] probs
  const float* img = (const float*)d_in[1];   // [1,3,96,96] rgb

  // Workspace layout (floats): feat[N*8] | sq[N] | sval[N] | partials[72]
  float* wsf      = (float*)d_ws;
  float* feat     = wsf;
  float* sq       = wsf + kN * 8;
  float* sval     = wsf + kN * 9;
  float* partials = wsf + kN * 10;

  crf_prep<<<(kN + 255) / 256, 256, 0, stream>>>(s, img, feat, sq, sval);
  crf_tiles<<<kBlocks, 256, 0, stream>>>(feat, sq, sval, partials);
  crf_finish<<<1, 32, 0, stream>>>(partials, (float*)d_out);
}